// GNN_80479097192825
// MI455X (gfx1250) — compile-verified
//
#include <hip/hip_runtime.h>

#define NN 100000
#define NE 1600000
#define D 128
#define D4 132          // padded LDS row stride (floats): 528B = 16B-aligned, bank-rotating
#define NL 7
#define GEMM_BLOCKS 1250  // 6250 row tiles / 1250 = 5 per block

typedef __attribute__((ext_vector_type(2))) float v2f;
typedef __attribute__((ext_vector_type(8))) float v8f;

// ---------------- CSR build (once per launch; dst is layer-invariant) ----------------

__global__ void gnn_hist_kernel(const int* __restrict__ dst, int* __restrict__ cnt) {
    int e = blockIdx.x * blockDim.x + threadIdx.x;
    if (e < NE) atomicAdd(&cnt[dst[e]], 1);
}

__global__ void gnn_scan_kernel(const int* __restrict__ cnt, int* __restrict__ row_ptr,
                                float* __restrict__ deg_inv) {
    __shared__ int sdata[1024];
    __shared__ int s_off;
    int tid = threadIdx.x;
    if (tid == 0) { s_off = 0; row_ptr[0] = 0; }
    __syncthreads();
    for (int base = 0; base < NN; base += 1024) {
        int i = base + tid;
        int v = (i < NN) ? cnt[i] : 0;
        sdata[tid] = v;
        __syncthreads();
        for (int off = 1; off < 1024; off <<= 1) {
            int t = (tid >= off) ? sdata[tid - off] : 0;
            __syncthreads();
            sdata[tid] += t;
            __syncthreads();
        }
        int incl = sdata[tid];
        int carry = s_off;
        if (i < NN) {
            row_ptr[i + 1] = carry + incl;
            deg_inv[i] = 1.0f / fmaxf((float)v, 1.0f);
        }
        __syncthreads();
        if (tid == 1023) s_off = carry + sdata[1023];
        __syncthreads();
    }
}

__global__ void gnn_fill_kernel(const int* __restrict__ src, const int* __restrict__ dst,
                                const int* __restrict__ row_ptr, int* __restrict__ cnt2,
                                int* __restrict__ col_src) {
    int e = blockIdx.x * blockDim.x + threadIdx.x;
    if (e < NE) {
        int d = dst[e];
        int pos = row_ptr[d] + atomicAdd(&cnt2[d], 1);
        col_src[pos] = src[e];
    }
}

// ---------------- Mean aggregation: one wave per dst node, float4 per lane ----------------

__global__ void __launch_bounds__(256)
gnn_agg_kernel(const float* __restrict__ h, const int* __restrict__ row_ptr,
               const int* __restrict__ col_src, const float* __restrict__ deg_inv,
               float* __restrict__ mean) {
    int wave = threadIdx.x >> 5;
    int lane = threadIdx.x & 31;
    int node = blockIdx.x * (blockDim.x >> 5) + wave;
    if (node >= NN) return;
    int e0 = row_ptr[node];
    int e1 = row_ptr[node + 1];
    float4 acc = make_float4(0.f, 0.f, 0.f, 0.f);
    for (int e = e0; e < e1; ++e) {
        int s = col_src[e];
        if (e + 1 < e1)
            __builtin_prefetch(h + (size_t)col_src[e + 1] * D + lane * 4, 0, 1);
        float4 v = ((const float4*)(h + (size_t)s * D))[lane];
        acc.x += v.x; acc.y += v.y; acc.z += v.z; acc.w += v.w;
    }
    float di = deg_inv[node];
    float4 r = make_float4(acc.x * di, acc.y * di, acc.z * di, acc.w * di);
    ((float4*)(mean + (size_t)node * D))[lane] = r;
}

// ---------------- Fused dual-GEMM + bias + ReLU via V_WMMA_F32_16X16X4_F32 ----------------
// Block = 8 waves; wave w owns col-tile w with its full B slab (Wl+Wr) in 128 VGPRs.
// A tiles (mean + x, 16 rows) staged in padded LDS, shared by all 8 waves.

__global__ void __launch_bounds__(256, 1)
gnn_gemm_kernel(const float* __restrict__ mean, const float* __restrict__ x,
                const float* __restrict__ Wl, const float* __restrict__ Wr,
                const float* __restrict__ bias, float* __restrict__ out, int relu) {
    __shared__ float sA[16 * D4];   // mean tile (padded rows)
    __shared__ float sX[16 * D4];   // x tile (padded rows)

    int wave = threadIdx.x >> 5;    // == col_tile (0..7)
    int lane = threadIdx.x & 31;
    int r     = lane & 15;
    int khalf = lane >> 4;

    // Preload this wave's B fragments for all 128 k (once per block).
    v2f bl[32], br[32];
    {
        const float* wlrow = Wl + (size_t)(wave * 16 + r) * D;
        const float* wrrow = Wr + (size_t)(wave * 16 + r) * D;
        #pragma unroll
        for (int kk = 0; kk < 32; ++kk) {
            int ko = kk * 4 + 2 * khalf;
            bl[kk] = *(const v2f*)(wlrow + ko);
            br[kk] = *(const v2f*)(wrrow + ko);
        }
    }
    float bj = bias[wave * 16 + r];

    for (int rt = blockIdx.x; rt < NN / 16; rt += GEMM_BLOCKS) {
        // Cooperative global->LDS copy of the 16x128 A tiles (float4, padded rows).
        {
            const float4* gm = (const float4*)(mean + (size_t)rt * 16 * D);
            const float4* gx = (const float4*)(x    + (size_t)rt * 16 * D);
            #pragma unroll
            for (int j = 0; j < 2; ++j) {
                int i    = threadIdx.x + j * 256;   // 0..511 float4 per matrix
                int row  = i >> 5;                  // 32 float4 per row
                int col4 = i & 31;
                ((float4*)(sA + row * D4))[col4] = gm[i];
                ((float4*)(sX + row * D4))[col4] = gx[i];
            }
        }
        __syncthreads();

        v8f acc = {};
        #pragma unroll
        for (int kk = 0; kk < 32; ++kk) {
            int ko = kk * 4 + 2 * khalf;
            v2f a = *(const v2f*)(sA + r * D4 + ko);
            acc = __builtin_amdgcn_wmma_f32_16x16x4_f32(false, a, false, bl[kk],
                                                        (short)0, acc, false, false);
            v2f a2 = *(const v2f*)(sX + r * D4 + ko);
            acc = __builtin_amdgcn_wmma_f32_16x16x4_f32(false, a2, false, br[kk],
                                                        (short)0, acc, false, false);
        }

        float* obase = out + (size_t)(rt * 16) * D + wave * 16 + r;
        #pragma unroll
        for (int v = 0; v < 8; ++v) {
            float val = acc[v] + bj;               // C layout: VGPR v = row v + 8*khalf
            if (relu) val = fmaxf(val, 0.0f);
            obase[(size_t)(v + 8 * khalf) * D] = val;
        }
        __syncthreads();   // A tiles reused next iteration
    }
}

// ---------------- Host orchestration (graph-capture safe) ----------------

extern "C" void kernel_launch(void* const* d_in, const int* in_sizes, int n_in,
                              void* d_out, int out_size, void* d_ws, size_t ws_size,
                              hipStream_t stream) {
    const float* x          = (const float*)d_in[0];
    const int*   edge_index = (const int*)d_in[1];   // (2, E): row0=src, row1=dst
    const float* Wl         = (const float*)d_in[2];
    const float* Wr         = (const float*)d_in[3];
    const float* b          = (const float*)d_in[4];
    const int* src = edge_index;
    const int* dst = edge_index + NE;

    char* ws = (char*)d_ws;
    size_t off = 0;
    auto carve = [&](size_t bytes) -> void* {
        void* p = ws + off;
        off = (off + bytes + 255) & ~(size_t)255;
        return p;
    };
    float* mean    = (float*)carve((size_t)NN * D * sizeof(float));
    float* hA      = (float*)carve((size_t)NN * D * sizeof(float));
    float* hB      = (float*)carve((size_t)NN * D * sizeof(float));
    float* deg_inv = (float*)carve((size_t)NN * sizeof(float));
    int*   row_ptr = (int*)carve((size_t)(NN + 1) * sizeof(int));
    int*   cnt     = (int*)carve((size_t)NN * sizeof(int));
    int*   col_src = (int*)carve((size_t)NE * sizeof(int));

    // CSR build (dst does not change across layers)
    hipMemsetAsync(cnt, 0, (size_t)NN * sizeof(int), stream);
    gnn_hist_kernel<<<(NE + 255) / 256, 256, 0, stream>>>(dst, cnt);
    gnn_scan_kernel<<<1, 1024, 0, stream>>>(cnt, row_ptr, deg_inv);
    hipMemsetAsync(cnt, 0, (size_t)NN * sizeof(int), stream);
    gnn_fill_kernel<<<(NE + 255) / 256, 256, 0, stream>>>(src, dst, row_ptr, cnt, col_src);

    // 7 layers, ping-pong hA/hB, final layer writes d_out
    const float* hin = x;
    float* houts[2] = {hA, hB};
    for (int l = 0; l < NL; ++l) {
        float* hout = (l == NL - 1) ? (float*)d_out : houts[l & 1];
        gnn_agg_kernel<<<NN / 8, 256, 0, stream>>>(hin, row_ptr, col_src, deg_inv, mean);
        gnn_gemm_kernel<<<GEMM_BLOCKS, 256, 0, stream>>>(
            mean, hin, Wl + (size_t)l * D * D, Wr + (size_t)l * D * D,
            b + (size_t)l * D, hout, (l < NL - 1) ? 1 : 0);
        hin = hout;
    }
}